// CompactnessLoss_7507602833894
// MI455X (gfx1250) — compile-verified
//
#include <hip/hip_runtime.h>

typedef float v2f __attribute__((ext_vector_type(2)));
typedef float v8f __attribute__((ext_vector_type(8)));

#define KCLUS 64
#define DIM   256
#define MAX_PART 256

// D = A(16x4 f32) * B(4x16 f32) + C, exact f32 — V_WMMA_F32_16X16X4_F32
__device__ __forceinline__ v8f wmma4(v2f a, v2f b, v8f c) {
  return __builtin_amdgcn_wmma_f32_16x16x4_f32(false, a, false, b, (short)0, c,
                                               false, false);
}

// ---------------- workspace layout (floats) ----------------
// [0..64)    sumsq_g   per-cluster sum of ||x||^2
// [64..128)  counts_g  (as uint32)
// [128..192) normsq_g  per-cluster ||sum_c||^2
// [1024.. )  part_sums: NBLK blocks of [64][256] f32 partial cluster sums
#define WS_PART_OFF 1024

__global__ void zero_small_ws(float* ws) {
  if (threadIdx.x < 192) ws[threadIdx.x] = 0.0f;
}

__global__ void count_pass(const int* __restrict__ assign, unsigned* __restrict__ counts_g, int N) {
  __shared__ unsigned h[KCLUS];
  if (threadIdx.x < KCLUS) h[threadIdx.x] = 0u;
  __syncthreads();
  int stride = gridDim.x * blockDim.x;
  for (int i = blockIdx.x * blockDim.x + threadIdx.x; i < N; i += stride) {
    int c = assign[i];
    if (c >= 0 && c < KCLUS) atomicAdd(&h[c], 1u);
  }
  __syncthreads();
  if (threadIdx.x < KCLUS) atomicAdd(&counts_g[threadIdx.x], h[threadIdx.x]);
}

// Main pass: 256 threads = 8 waves. Wave w owns columns [32w, 32w+32).
// Both the segmented sum AND the segmented sum-of-squares go through
// V_WMMA_F32_16X16X4_F32 with shared one-hot A fragments (exact f32).
__global__ void __launch_bounds__(256)
main_pass(const float* __restrict__ F, const int* __restrict__ assign,
          float* __restrict__ part_sums, float* __restrict__ sumsq_g,
          int N, int rowsPerBlock) {
  const int tid  = threadIdx.x;
  const int wave = tid >> 5;
  const int lane = tid & 31;
  const int half = lane >> 4;   // upper/lower 16-lane group
  const int l16  = lane & 15;
  const int colBase = wave * 32;
  const int c0 = colBase + l16;

  __shared__ float lds_sumsq[KCLUS];
  if (tid < KCLUS) lds_sumsq[tid] = 0.0f;
  __syncthreads();

  // 8 sum tiles [m][t] + 4 sumsq tiles [m]  (m = cluster tile, t = col tile)
  v8f a00 = {}, a01 = {}, a10 = {}, a11 = {}, a20 = {}, a21 = {}, a30 = {}, a31 = {};
  v8f q0 = {}, q1 = {}, q2 = {}, q3 = {};

  const int rowStart = blockIdx.x * rowsPerBlock;
  const int rowEnd   = min(rowStart + rowsPerBlock, N);
  const int nRows    = rowEnd > rowStart ? rowEnd - rowStart : 0;
  const int fullEnd  = rowStart + (nRows & ~3);   // 4-row-aligned main range

  // ---------------- main loop: no guards, no LDS, no shuffles ----------------
  const float* pA = F + (size_t)(rowStart + 2 * half) * DIM + c0;
  for (int r0 = rowStart; r0 < fullEnd; r0 += 4, pA += 4 * DIM) {
    // Uniform scalar loads of the 4 assignments for this slab (s_load path).
    const int s0 = assign[r0];
    const int s1 = assign[r0 + 1];
    const int s2 = assign[r0 + 2];
    const int s3 = assign[r0 + 3];
    const int cA = half ? s2 : s0;    // cluster of this lane's K row rA
    const int cB = half ? s3 : s1;    // cluster of this lane's K row rB

    v2f b0, b1;                       // col tiles t=0 (cols c0) and t=1 (cols c0+16)
    b0.x = pA[0];
    b0.y = pA[DIM];
    b1.x = pA[16];
    b1.y = pA[DIM + 16];

    // speculative prefetch of the next 4-row slab (covers this wave's columns)
    __builtin_prefetch(pA + 4 * DIM, 0, 0);
    __builtin_prefetch(pA + 5 * DIM, 0, 0);

    v2f bq;                           // per-row x^2 partial over this wave's 32 cols
    bq.x = __builtin_fmaf(b0.x, b0.x, b1.x * b1.x);
    bq.y = __builtin_fmaf(b0.y, b0.y, b1.y * b1.y);

    v2f a;
    a.x = (cA == l16) ? 1.0f : 0.0f;
    a.y = (cB == l16) ? 1.0f : 0.0f;
    a00 = wmma4(a, b0, a00); a01 = wmma4(a, b1, a01); q0 = wmma4(a, bq, q0);
    a.x = (cA == 16 + l16) ? 1.0f : 0.0f;
    a.y = (cB == 16 + l16) ? 1.0f : 0.0f;
    a10 = wmma4(a, b0, a10); a11 = wmma4(a, b1, a11); q1 = wmma4(a, bq, q1);
    a.x = (cA == 32 + l16) ? 1.0f : 0.0f;
    a.y = (cB == 32 + l16) ? 1.0f : 0.0f;
    a20 = wmma4(a, b0, a20); a21 = wmma4(a, b1, a21); q2 = wmma4(a, bq, q2);
    a.x = (cA == 48 + l16) ? 1.0f : 0.0f;
    a.y = (cB == 48 + l16) ? 1.0f : 0.0f;
    a30 = wmma4(a, b0, a30); a31 = wmma4(a, b1, a31); q3 = wmma4(a, bq, q3);
  }

  // ---------------- tail loop (< 4 rows): guarded, same WMMA body ----------------
  for (int r0 = fullEnd; r0 < rowEnd; r0 += 4) {
    const int rA = r0 + 2 * half;
    const int rB = rA + 1;
    const bool vA = (rA < rowEnd);
    const bool vB = (rB < rowEnd);

    const int cA = vA ? assign[rA] : -1;
    const int cB = vB ? assign[rB] : -1;

    const float* rowPA = F + (size_t)rA * DIM;
    const float* rowPB = F + (size_t)rB * DIM;

    v2f b0, b1;
    b0.x = vA ? rowPA[c0]      : 0.0f;
    b0.y = vB ? rowPB[c0]      : 0.0f;
    b1.x = vA ? rowPA[c0 + 16] : 0.0f;
    b1.y = vB ? rowPB[c0 + 16] : 0.0f;

    v2f bq;
    bq.x = __builtin_fmaf(b0.x, b0.x, b1.x * b1.x);
    bq.y = __builtin_fmaf(b0.y, b0.y, b1.y * b1.y);

    v2f a;
    a.x = (cA == l16) ? 1.0f : 0.0f;
    a.y = (cB == l16) ? 1.0f : 0.0f;
    a00 = wmma4(a, b0, a00); a01 = wmma4(a, b1, a01); q0 = wmma4(a, bq, q0);
    a.x = (cA == 16 + l16) ? 1.0f : 0.0f;
    a.y = (cB == 16 + l16) ? 1.0f : 0.0f;
    a10 = wmma4(a, b0, a10); a11 = wmma4(a, b1, a11); q1 = wmma4(a, bq, q1);
    a.x = (cA == 32 + l16) ? 1.0f : 0.0f;
    a.y = (cB == 32 + l16) ? 1.0f : 0.0f;
    a20 = wmma4(a, b0, a20); a21 = wmma4(a, b1, a21); q2 = wmma4(a, bq, q2);
    a.x = (cA == 48 + l16) ? 1.0f : 0.0f;
    a.y = (cB == 48 + l16) ? 1.0f : 0.0f;
    a30 = wmma4(a, b0, a30); a31 = wmma4(a, b1, a31); q3 = wmma4(a, bq, q3);
  }

  // Flush sum tiles. C/D layout: VGPR v -> row M = v + 8*half; lane l16 -> col.
  float* base = part_sums + (size_t)blockIdx.x * (KCLUS * DIM);
  #pragma unroll
  for (int m = 0; m < 4; ++m) {
    const v8f* t0 = (m == 0) ? &a00 : (m == 1) ? &a10 : (m == 2) ? &a20 : &a30;
    const v8f* t1 = (m == 0) ? &a01 : (m == 1) ? &a11 : (m == 2) ? &a21 : &a31;
    #pragma unroll
    for (int v = 0; v < 8; ++v) {
      const int clus = m * 16 + half * 8 + v;
      base[clus * DIM + colBase + l16]      = (*t0)[v];
      base[clus * DIM + colBase + 16 + l16] = (*t1)[v];
    }
  }

  // Flush sumsq tiles into the per-cluster LDS accumulator (once per block).
  #pragma unroll
  for (int m = 0; m < 4; ++m) {
    const v8f* q = (m == 0) ? &q0 : (m == 1) ? &q1 : (m == 2) ? &q2 : &q3;
    #pragma unroll
    for (int v = 0; v < 8; ++v) {
      atomicAdd(&lds_sumsq[m * 16 + half * 8 + v], (*q)[v]);
    }
  }

  __syncthreads();
  if (tid < KCLUS) atomicAdd(&sumsq_g[tid], lds_sumsq[tid]);
}

// One block per cluster: fold NBLK partials, produce ||sum_c||^2
__global__ void __launch_bounds__(256)
reduce_pass(const float* __restrict__ part_sums, float* __restrict__ normsq_g, int nblk) {
  const int col  = threadIdx.x;       // 256 threads = DIM
  const int clus = blockIdx.x;
  float s = 0.0f;
  const float* p = part_sums + clus * DIM + col;
  for (int b = 0; b < nblk; ++b) s += p[(size_t)b * (KCLUS * DIM)];
  __shared__ float red[DIM];
  red[col] = s * s;
  __syncthreads();
  for (int o = DIM / 2; o > 0; o >>= 1) {
    if (col < o) red[col] += red[col + o];
    __syncthreads();
  }
  if (col == 0) normsq_g[clus] = red[0];
}

__global__ void final_pass(const float* __restrict__ sumsq_g,
                           const unsigned* __restrict__ counts_g,
                           const float* __restrict__ normsq_g,
                           float* __restrict__ out) {
  const int tid = threadIdx.x;
  __shared__ float red[KCLUS];
  float v = 0.0f;
  if (tid < KCLUS) {
    unsigned n = counts_g[tid];
    if (n > 1u) v = sumsq_g[tid] - normsq_g[tid] / (float)n;
  }
  red[tid] = v;
  __syncthreads();
  for (int o = KCLUS / 2; o > 0; o >>= 1) {
    if (tid < o) red[tid] += red[tid + o];
    __syncthreads();
  }
  if (tid == 0) out[0] = red[0];
}

extern "C" void kernel_launch(void* const* d_in, const int* in_sizes, int n_in,
                              void* d_out, int out_size, void* d_ws, size_t ws_size,
                              hipStream_t stream) {
  const float* F      = (const float*)d_in[0];
  const int*   assign = (const int*)d_in[1];
  const int N = in_sizes[1];

  float*    ws_f     = (float*)d_ws;
  float*    sumsq_g  = ws_f;
  unsigned* counts_g = (unsigned*)(ws_f + 64);
  float*    normsq_g = ws_f + 128;
  float*    part     = ws_f + WS_PART_OFF;

  // Adapt partial-buffer count to available workspace (deterministic per run).
  size_t avail = (ws_size > WS_PART_OFF * sizeof(float))
                     ? ws_size - WS_PART_OFF * sizeof(float) : 0;
  int nblk = (int)(avail / ((size_t)KCLUS * DIM * sizeof(float)));
  if (nblk < 1) nblk = 1;
  if (nblk > MAX_PART) nblk = MAX_PART;
  int rowsPerBlock = (N + nblk - 1) / nblk;

  zero_small_ws<<<1, 256, 0, stream>>>(ws_f);
  count_pass<<<128, 256, 0, stream>>>(assign, counts_g, N);
  main_pass<<<nblk, 256, 0, stream>>>(F, assign, part, sumsq_g, N, rowsPerBlock);
  reduce_pass<<<KCLUS, DIM, 0, stream>>>(part, normsq_g, nblk);
  final_pass<<<1, KCLUS, 0, stream>>>(sumsq_g, counts_g, normsq_g, (float*)d_out);
}